// VerificationLayer_58282706207345
// MI455X (gfx1250) — compile-verified
//
#include <hip/hip_runtime.h>
#include <hip/hip_bf16.h>
#include <math.h>

typedef __attribute__((ext_vector_type(2))) float    v2f;
typedef __attribute__((ext_vector_type(8))) float    v8f;
typedef __attribute__((ext_vector_type(4))) uint32_t v4u;
typedef __attribute__((ext_vector_type(8))) uint32_t v8u;

#define KTOT 1024
#define NDIM 512
#define ROWSTRIDE 36     // 32 data DW + 4 pad DW (TDM pad) -> 36 % 32 == 4: conflict-free banks

static __device__ __forceinline__ uint32_t rfl(uint32_t x) {
    return (uint32_t)__builtin_amdgcn_readfirstlane((int)x);
}

// generic LDS pointer = {shared_aperture_hi32, lds_byte_offset} -> truncate for D#.lds_addr
static __device__ __forceinline__ uint32_t lds_off(const void* p) {
    return (uint32_t)(uintptr_t)p;
}

// Issue one TDM 2D tile load: 16 rows (stride 1 MB) x 32 f32, padded into LDS.
static __device__ __forceinline__ void tdm_issue(uint32_t galo, uint32_t gahi,
                                                 uint32_t ldsOff, v8u g1) {
    v4u g0;
    g0.x = 1u;                                   // count=1 (valid user descriptor)
    g0.y = ldsOff;                               // lds_addr (bytes)
    g0.z = galo;                                 // global_addr[31:0]
    g0.w = (gahi & 0x01FFFFFFu) | 0x80000000u;   // global_addr[56:32] | type=2<<30
    asm volatile("tensor_load_to_lds %0, %1" :: "s"(g0), "s"(g1) : "memory");
}

// grid.x = 64 k-groups (16 k each), grid.y = 8 i-slices (64 i each); 256 thr = 8 waves
__global__ __launch_bounds__(256) void quad_tdm_wmma_kernel(const float* __restrict__ s,
                                                            const float* __restrict__ P,
                                                            float* __restrict__ partial) {
    __shared__ float s_s[NDIM];
    __shared__ float bufs[8][2][16 * ROWSTRIDE];   // per-wave double buffers (36 KB)
    __shared__ float smw[8 * 16];

    const int tid  = threadIdx.x;
    const int lane = tid & 31;
    const int wv   = tid >> 5;          // wave id 0..7 (wave32)
    const int hi   = lane >> 4;
    const int lo   = lane & 15;
    const int k0   = blockIdx.x * 16;
    const int i0   = blockIdx.y * 64;

    s_s[tid]       = s[tid];
    s_s[tid + 256] = s[tid + 256];

    // D# group 1 (constant): mask=0 | data_size=4B | pad_en | interval=32DW | amount=4DW
    v8u g1;
    g1[0] = (2u << 16) | (1u << 20) | (4u << 22) | (3u << 25);  // 0x07120000
    g1[1] = 32u << 16;        // tensor_dim0 = 32
    g1[2] = 16u << 16;        // tensor_dim1 = 16
    g1[3] = 32u << 16;        // tile_dim0 = 32
    g1[4] = 16u;              // tile_dim1 = 16
    g1[5] = (uint32_t)(NDIM * NDIM);   // tensor_dim0_stride = 262144 elements
    g1[6] = 0u;
    g1[7] = 0u;

    const uint32_t bufOff0 = rfl(lds_off(&bufs[wv][0][0]));
    const uint32_t bufOff1 = rfl(lds_off(&bufs[wv][1][0]));

    const uint64_t pbase = (uint64_t)(uintptr_t)P;
    const uint32_t jwave = (uint32_t)(wv * 32);    // this wave's 32-col slice within a half

    // step = 2*(t - i0) + half ; half selects columns [half*256, half*256+256)
    auto issue = [&](int stp) {
        const int t    = i0 + (stp >> 1);
        const int half = stp & 1;
        const uint64_t ga = pbase +
            (((uint64_t)k0 * (uint64_t)(NDIM * NDIM)) +
             (uint64_t)t * (uint64_t)NDIM +
             (uint64_t)(half * 256) + (uint64_t)jwave) * 4ull;
        tdm_issue(rfl((uint32_t)ga), rfl((uint32_t)(ga >> 32)),
                  (stp & 1) ? bufOff1 : bufOff0, g1);
    };

    __syncthreads();           // s_s visible to all waves

    issue(0);                  // prime the pipeline

    v8f acc = {0.f, 0.f, 0.f, 0.f, 0.f, 0.f, 0.f, 0.f};

    for (int step = 0; step < 128; ++step) {
        if (step < 127) {
            issue(step + 1);
            __builtin_amdgcn_s_wait_tensorcnt(1);   // step's DMA done (in-order per wave)
        } else {
            __builtin_amdgcn_s_wait_tensorcnt(0);
        }

        const float* bp = &bufs[wv][step & 1][0];
        const float  st = s_s[i0 + (step >> 1)];                 // s[i]
        const int    jg = ((step & 1) << 8) + wv * 32;           // global j base of slice

#pragma unroll
        for (int c0 = 0; c0 < 32; c0 += 4) {
            const int cl = c0 + 2 * hi;
            // A fragment: lane = (row lo, cols cl, cl+1) of the padded tile
            v2f af = *(const v2f*)&bp[lo * ROWSTRIDE + cl];
            // B fragment: replicated over N -> depends only on K row c
            v2f sf = *(const v2f*)&s_s[jg + cl];
            v2f bf = sf * st;                                    // s[i] * s[j]
            acc = __builtin_amdgcn_wmma_f32_16x16x4_f32(
                false, af, false, bf, (short)0, acc, false, false);
        }
    }

    // D column N=0 lives in lanes 0 and 16: lane L, vgpr r -> kk = r + 8*(L>>4)
    if (lo == 0) {
#pragma unroll
        for (int r = 0; r < 8; ++r)
            smw[wv * 16 + r + 8 * hi] = acc[r];
    }
    __syncthreads();

    if (tid < 16) {
        float sum = 0.f;
#pragma unroll
        for (int w = 0; w < 8; ++w) sum += smw[w * 16 + tid];
        partial[(blockIdx.x * 8 + blockIdx.y) * 16 + tid] = sum;
    }
}

// finalize: quad -> round(sin^2) -> sum -> relu(1 - total)
__global__ __launch_bounds__(256) void finalize_kernel(const float* __restrict__ m,
                                                       const float* __restrict__ partial,
                                                       float* __restrict__ out) {
    __shared__ float red[256];
    const int tid = threadIdx.x;
    float local = 0.f;
#pragma unroll
    for (int k = tid; k < KTOT; k += 256) {
        const int kg = k >> 4;
        const int kk = k & 15;
        float q = 0.f;
#pragma unroll
        for (int iy = 0; iy < 8; ++iy)
            q += partial[(kg * 8 + iy) * 16 + kk];
        const float x  = q - m[k];
        const float sv = sinf(x * 1.57079632679489662f);
        const float y  = sv * sv;
        local += rintf(y);                  // round-half-even, matches jnp.round
    }
    red[tid] = local;
    __syncthreads();
    for (int off = 128; off > 0; off >>= 1) {
        if (tid < off) red[tid] += red[tid + off];
        __syncthreads();
    }
    if (tid == 0) out[0] = fmaxf(0.f, 1.f - red[0]);
}

extern "C" void kernel_launch(void* const* d_in, const int* in_sizes, int n_in,
                              void* d_out, int out_size, void* d_ws, size_t ws_size,
                              hipStream_t stream) {
    const float* m = (const float*)d_in[0];   // (1024,1) f32
    const float* s = (const float*)d_in[1];   // (512,1)  f32
    const float* P = (const float*)d_in[2];   // (1024,512,512) f32
    float* out     = (float*)d_out;           // scalar f32
    float* partial = (float*)d_ws;            // 64*8*16 = 8192 floats (32 KB)

    (void)in_sizes; (void)n_in; (void)out_size; (void)ws_size;

    dim3 grid(KTOT / 16, 8);
    quad_tdm_wmma_kernel<<<grid, 256, 0, stream>>>(s, P, partial);
    finalize_kernel<<<1, 256, 0, stream>>>(m, partial, out);
}